// Elbo_88330297410273
// MI455X (gfx1250) — compile-verified
//
#include <hip/hip_runtime.h>
#include <math.h>

#define B_  16
#define H_  448
#define W_  1024
#define HW_ (H_ * W_)            // 458752
#define N_  (B_ * HW_)           // 7340032
#define Q_  (N_ / 4)             // 1835008 quads
#define QHW_ (HW_ / 4)           // 114688 quads per batch image

// exp(x/2) = exp2(x * 0.5*log2(e))
#define K_HALF_LOG2E 0.72134752044448169764f

__device__ __forceinline__ float4 ld4(const float* __restrict__ p) {
    return *reinterpret_cast<const float4*>(p);
}

__global__ void zero_acc_kernel(double* __restrict__ acc) {
    if (threadIdx.x < 2) acc[threadIdx.x] = 0.0;
}

__global__ __launch_bounds__(256) void elbo_epe_kernel(
        const float* __restrict__ mean, const float* __restrict__ logv,
        const float* __restrict__ img1, const float* __restrict__ img2,
        const float* __restrict__ target, const float* __restrict__ eps,
        double* __restrict__ acc) {
    float a_elbo = 0.0f;
    float a_epe  = 0.0f;

    const int q = blockIdx.x * blockDim.x + threadIdx.x;
    if (q < Q_) {
        const int b  = q / QHW_;                 // one magic-multiply, once per thread
        const int r  = (q - b * QHW_) * 4;       // pixel index within the (H,W) image
        const int y  = r >> 10;                  // W_ == 1024
        const int x  = r & (W_ - 1);             // multiple of 4

        const float* mb = mean   + b * 2 * HW_;
        const float* lb = logv   + b * 2 * HW_;
        const float* eb = eps    + b * 2 * HW_;
        const float* i1 = img1   + b * 3 * HW_;
        const float* i2 = img2   + b * 3 * HW_;
        const float* tb = target + b * 2 * HW_;

        const bool has_down  = (y < H_ - 1);
        const bool has_right = (x + 4 < W_);     // pixel x+3's forward-diff neighbor exists

        // Software-pipeline hint: start pulling the down-row flow params now
        // (consumed after the gather-heavy data term).  gfx1250 global_prefetch_b8.
        if (has_down) {
            __builtin_prefetch(mb + r + W_, 0, 0);
            __builtin_prefetch(lb + r + W_, 0, 0);
            __builtin_prefetch(eb + r + W_, 0, 0);
        }

        // --- center flow sample for 4 consecutive pixels (b128 loads) ---
        const float4 m0v = ld4(mb + r);
        const float4 m1v = ld4(mb + HW_ + r);
        const float4 l0v = ld4(lb + r);
        const float4 l1v = ld4(lb + HW_ + r);
        const float4 e0v = ld4(eb + r);
        const float4 e1v = ld4(eb + HW_ + r);

        float m0[4] = {m0v.x, m0v.y, m0v.z, m0v.w};
        float m1[4] = {m1v.x, m1v.y, m1v.z, m1v.w};
        float l0[4] = {l0v.x, l0v.y, l0v.z, l0v.w};
        float l1[4] = {l1v.x, l1v.y, l1v.z, l1v.w};
        float e0[4] = {e0v.x, e0v.y, e0v.z, e0v.w};
        float e1[4] = {e1v.x, e1v.y, e1v.z, e1v.w};

        float u[4], v[4];
#pragma unroll
        for (int i = 0; i < 4; ++i) {
            u[i] = fmaf(exp2f(l0[i] * K_HALF_LOG2E), e0[i], m0[i]);
            v[i] = fmaf(exp2f(l1[i] * K_HALF_LOG2E), e1[i], m1[i]);
        }

        // --- data term: bilinear warp of img2, clamped borders ---
        const float4 i1r = ld4(i1 + r);
        const float4 i1g = ld4(i1 + HW_ + r);
        const float4 i1b = ld4(i1 + 2 * HW_ + r);
        const float c1[3][4] = {{i1r.x, i1r.y, i1r.z, i1r.w},
                                {i1g.x, i1g.y, i1g.z, i1g.w},
                                {i1b.x, i1b.y, i1b.z, i1b.w}};
#pragma unroll
        for (int i = 0; i < 4; ++i) {
            const float fx  = (float)(x + i) + u[i];
            const float fy  = (float)y + v[i];
            const float x0f = floorf(fx);
            const float y0f = floorf(fy);
            const float wx  = fx - x0f;
            const float wy  = fy - y0f;
            int x0 = (int)x0f; x0 = x0 < 0 ? 0 : (x0 > W_ - 1 ? W_ - 1 : x0);
            int y0 = (int)y0f; y0 = y0 < 0 ? 0 : (y0 > H_ - 1 ? H_ - 1 : y0);
            const int x1 = (x0 + 1 > W_ - 1) ? (W_ - 1) : x0 + 1;
            const int y1 = (y0 + 1 > H_ - 1) ? (H_ - 1) : y0 + 1;
            const int i00 = y0 * W_ + x0;
            const int i01 = y0 * W_ + x1;
            const int i10 = y1 * W_ + x0;
            const int i11 = y1 * W_ + x1;

            float A = 0.0f;
#pragma unroll
            for (int c = 0; c < 3; ++c) {
                const float* pc  = i2 + c * HW_;
                const float v00 = pc[i00];
                const float v01 = pc[i01];
                const float v10 = pc[i10];
                const float v11 = pc[i11];
                const float top = fmaf(wx, v01 - v00, v00);
                const float bot = fmaf(wx, v11 - v10, v10);
                const float w   = fmaf(wy, bot - top, top);
                const float d   = c1[c][i] - w;
                A = fmaf(d, d, A);
            }
            a_elbo += sqrtf(A + 1e-5f);
        }

        // --- down-row flow sample (b128 loads) for the vertical forward diff ---
        float ud[4], vd[4];
        if (has_down) {
            const float4 dm0 = ld4(mb + r + W_);
            const float4 dm1 = ld4(mb + HW_ + r + W_);
            const float4 dl0 = ld4(lb + r + W_);
            const float4 dl1 = ld4(lb + HW_ + r + W_);
            const float4 de0 = ld4(eb + r + W_);
            const float4 de1 = ld4(eb + HW_ + r + W_);
            const float a0[4] = {dm0.x, dm0.y, dm0.z, dm0.w};
            const float a1[4] = {dm1.x, dm1.y, dm1.z, dm1.w};
            const float b0[4] = {dl0.x, dl0.y, dl0.z, dl0.w};
            const float b1[4] = {dl1.x, dl1.y, dl1.z, dl1.w};
            const float g0[4] = {de0.x, de0.y, de0.z, de0.w};
            const float g1[4] = {de1.x, de1.y, de1.z, de1.w};
#pragma unroll
            for (int i = 0; i < 4; ++i) {
                ud[i] = fmaf(exp2f(b0[i] * K_HALF_LOG2E), g0[i], a0[i]);
                vd[i] = fmaf(exp2f(b1[i] * K_HALF_LOG2E), g1[i], a1[i]);
            }
        }

        // --- right-boundary flow sample (pixel x+4), scalar loads ---
        float ur = 0.0f, vr = 0.0f;
        if (has_right) {
            const int rn = r + 4;
            ur = fmaf(exp2f(lb[rn]       * K_HALF_LOG2E), eb[rn],       mb[rn]);
            vr = fmaf(exp2f(lb[HW_ + rn] * K_HALF_LOG2E), eb[HW_ + rn], mb[HW_ + rn]);
        }

        // --- smoothness penalty + log-var term ---
#pragma unroll
        for (int i = 0; i < 4; ++i) {
            float Bs = 0.0f;
            // horizontal forward diff: neighbor in-register for i<3
            if (i < 3) {
                const float du = u[i + 1] - u[i];
                const float dv = v[i + 1] - v[i];
                Bs = fmaf(du, du, fmaf(dv, dv, Bs));
            } else if (has_right) {
                const float du = ur - u[3];
                const float dv = vr - v[3];
                Bs = fmaf(du, du, fmaf(dv, dv, Bs));
            }
            if (has_down) {
                const float du = ud[i] - u[i];
                const float dv = vd[i] - v[i];
                Bs = fmaf(du, du, fmaf(dv, dv, Bs));
            }
            a_elbo += sqrtf(Bs + 1e-5f);
            a_elbo -= 0.5f * (l0[i] + l1[i]);
        }

        // --- EPE (uses mean, not the sampled flow), b128 loads ---
        const float4 t0v = ld4(tb + r);
        const float4 t1v = ld4(tb + HW_ + r);
        const float t0[4] = {t0v.x, t0v.y, t0v.z, t0v.w};
        const float t1[4] = {t1v.x, t1v.y, t1v.z, t1v.w};
#pragma unroll
        for (int i = 0; i < 4; ++i) {
            const float d0 = m0[i] - t0[i];
            const float d1 = m1[i] - t1[i];
            a_epe += sqrtf(fmaf(d0, d0, d1 * d1));
        }
    }

    // --- block reduction (8 wave32 waves) ---
    __shared__ float s0[256];
    __shared__ float s1[256];
    const int t = threadIdx.x;
    s0[t] = a_elbo;
    s1[t] = a_epe;
    __syncthreads();
#pragma unroll
    for (int off = 128; off > 0; off >>= 1) {
        if (t < off) {
            s0[t] += s0[t + off];
            s1[t] += s1[t + off];
        }
        __syncthreads();
    }
    if (t == 0) {
        atomicAdd(&acc[0], (double)s0[0]);
        atomicAdd(&acc[1], (double)s1[0]);
    }
}

__global__ void finalize_kernel(const double* __restrict__ acc, float* __restrict__ out) {
    if (threadIdx.x == 0) {
        out[0] = (float)(acc[0] / (double)B_);   // mean over batch
        out[1] = (float)(acc[1] / (double)N_);   // mean over B*H*W
    }
}

extern "C" void kernel_launch(void* const* d_in, const int* in_sizes, int n_in,
                              void* d_out, int out_size, void* d_ws, size_t ws_size,
                              hipStream_t stream) {
    (void)in_sizes; (void)n_in; (void)out_size; (void)ws_size;
    const float* mean   = (const float*)d_in[0];
    const float* logv   = (const float*)d_in[1];
    const float* img1   = (const float*)d_in[2];
    const float* img2   = (const float*)d_in[3];
    const float* target = (const float*)d_in[4];
    const float* eps    = (const float*)d_in[5];
    float*  out = (float*)d_out;
    double* acc = (double*)d_ws;   // 2 doubles of scratch

    zero_acc_kernel<<<1, 32, 0, stream>>>(acc);
    const int blocks = (Q_ + 255) / 256;         // 7168
    elbo_epe_kernel<<<blocks, 256, 0, stream>>>(mean, logv, img1, img2, target, eps, acc);
    finalize_kernel<<<1, 32, 0, stream>>>(acc, out);
}